// DaleLaw_25409026524030
// MI455X (gfx1250) — compile-verified
//
#include <hip/hip_runtime.h>

// DaleLaw loss: loss = sum( W * (t*W - (1-t)*sign(R)) ), t = 0.5, fp32 8192x8192.
// Memory-bound streaming reduction: 512 MiB read @ 23.3 TB/s => ~23 us floor.
// Strategy: NT b128 vector loads (no L2 pollution, zero reuse), scalar-counted
// uniform hot loop (no exec-mask churn), 2-FMA/element accumulation of
// sum(w^2 - w*sign(r)) with the t=0.5 scale hoisted out, wave reduction via
// V_WMMA_F32_16X16X4_F32 (B = ones => row sums), ds_swizzle lane swaps,
// LDS cross-wave combine, two-pass deterministic final reduce.

typedef float v2f __attribute__((ext_vector_type(2)));
typedef float v4f __attribute__((ext_vector_type(4)));
typedef float v8f __attribute__((ext_vector_type(8)));

#define T_COEF 0.5f

// XOR-lane swap via ds_swizzle_b32, group-of-32 encoding:
// offset = and_mask(0x1F) | or_mask<<5 (0) | xor_mask<<10, bit15 = 0.
// The offset must be an ICE -> template parameter.
template <int SWZ_OFF>
__device__ __forceinline__ float lane_xor_add(float x) {
  int other = __builtin_amdgcn_ds_swizzle(__float_as_int(x), SWZ_OFF);
  return x + __int_as_float(other);
}
#define SWZ_XOR16 0x401F
#define SWZ_XOR4  0x101F
#define SWZ_XOR2  0x081F
#define SWZ_XOR1  0x041F

// Wave32 reduction on the f32 matrix pipe.
// A (16x4 f32, 2 VGPRs): a0 = lane partial, a1 = 0.
//   VGPR0: lanes 0-15 -> K=0 rows M=0..15; lanes 16-31 -> K=2 rows M=0..15.
// B (4x16) = all ones, C = 0  =>  D[m][n] = acc[m] + acc[m+16] for every n.
// Lane L<16 holds D[0..7][L] in d0..d7; lane L>=16 holds D[8..15][L-16].
// s = sum(d0..d7) covers half the rows; XOR-16 swap finishes the wave sum.
// Requires EXEC == all ones: only called from fully converged code.
__device__ __forceinline__ float wave_reduce_wmma(float x) {
  v2f a; a.x = x;    a.y = 0.0f;
  v2f b; b.x = 1.0f; b.y = 1.0f;
  v8f c = {0.0f, 0.0f, 0.0f, 0.0f, 0.0f, 0.0f, 0.0f, 0.0f};
  v8f d = __builtin_amdgcn_wmma_f32_16x16x4_f32(
      /*neg_a=*/false, a, /*neg_b=*/false, b,
      /*c_mod=*/(short)0, c, /*reuse_a=*/false, /*reuse_b=*/false);
  float s = d[0] + d[1] + d[2] + d[3] + d[4] + d[5] + d[6] + d[7];
  return lane_xor_add<SWZ_XOR16>(s);  // full wave sum in every lane
}

// Per-element contribution to sum(w^2 - w*sign(r)); caller applies the 0.5.
__device__ __forceinline__ void accum_elem(float wj, float rj, float& acc) {
  float sg = (rj != 0.0f) ? __builtin_copysignf(1.0f, rj) : 0.0f;  // bfi+cmp+cndmask
  acc = fmaf(wj, wj, acc);    // + w^2
  acc = fmaf(-wj, sg, acc);   // - w*sign(r)   (neg modifier is free)
}

__device__ __forceinline__ void block_reduce_store(float comb, float* dst) {
  float wsum = wave_reduce_wmma(comb);

  __shared__ float lsum[8];
  const int lane = (int)(threadIdx.x & 31u);
  const int wid  = (int)(threadIdx.x >> 5u);
  if (lane == 0) lsum[wid] = wsum;
  __syncthreads();

  if (wid == 0) {
    float v = (lane < 8) ? lsum[lane] : 0.0f;
    v = lane_xor_add<SWZ_XOR4>(v);
    v = lane_xor_add<SWZ_XOR2>(v);
    v = lane_xor_add<SWZ_XOR1>(v);
    if (lane == 0) *dst = v;
  }
}

__global__ __launch_bounds__(256) void dale_partial_kernel(
    const float* __restrict__ w, const float* __restrict__ r,
    float* __restrict__ partials, int nv4, int niter) {
  const v4f* __restrict__ w4 = (const v4f*)w;
  const v4f* __restrict__ r4 = (const v4f*)r;

  const int tid    = (int)(blockIdx.x * 256u + threadIdx.x);
  const int stride = (int)(gridDim.x * 256u);

  float acc = 0.0f;
  // Uniform scalar trip count (niter = nv4/stride = 32 here): loop control is
  // SALU-only, EXEC stays all-ones throughout -- legal for the WMMA below.
#pragma unroll 4
  for (int k = 0; k < niter; ++k) {
    const int i = tid + k * stride;
    v4f wv = __builtin_nontemporal_load(&w4[i]);  // global_load_b128 NT
    v4f rv = __builtin_nontemporal_load(&r4[i]);
#pragma unroll
    for (int j = 0; j < 4; ++j) accum_elem(wv[j], rv[j], acc);
  }
  // Tail for non-divisible sizes (empty for 8192^2): reconverges before WMMA.
  {
    const int i = tid + niter * stride;
    if (i < nv4) {
      v4f wv = __builtin_nontemporal_load(&w4[i]);
      v4f rv = __builtin_nontemporal_load(&r4[i]);
#pragma unroll
      for (int j = 0; j < 4; ++j) accum_elem(wv[j], rv[j], acc);
    }
  }

  // Apply t = 0.5 once per thread: loss = 0.5 * sum(w^2 - w*sign(r)).
  block_reduce_store(T_COEF * acc, &partials[blockIdx.x]);
}

__global__ __launch_bounds__(256) void dale_final_kernel(
    const float* __restrict__ partials, float* __restrict__ out, int n) {
  float acc = 0.0f;
  // n = 2048, 256 threads -> exactly 8 uniform iterations.
  for (int i = (int)threadIdx.x; i < n; i += 256) acc += partials[i];
  block_reduce_store(acc, out);
}

extern "C" void kernel_launch(void* const* d_in, const int* in_sizes, int n_in,
                              void* d_out, int out_size, void* d_ws, size_t ws_size,
                              hipStream_t stream) {
  const float* w = (const float*)d_in[0];  // weights
  const float* r = (const float*)d_in[1];  // reference_weights
  float* out      = (float*)d_out;
  float* partials = (float*)d_ws;          // 2048 floats of scratch

  const int n_elems = in_sizes[0];         // 8192*8192 = 67,108,864
  const int nv4     = n_elems >> 2;        // 16,777,216 float4 pairs

  const int kBlocks = 2048;
  const int stride  = kBlocks * 256;       // 524,288 threads
  const int niter   = nv4 / stride;        // exactly 32, no tail

  dale_partial_kernel<<<kBlocks, 256, 0, stream>>>(w, r, partials, nv4, niter);
  dale_final_kernel<<<1, 256, 0, stream>>>(partials, out, kBlocks);
}